// Light_Layer_79774722556236
// MI455X (gfx1250) — compile-verified
//
#include <hip/hip_runtime.h>
#include <hip/hip_bf16.h>
#include <cstdint>

#define TILE 256  // edges staged per block

typedef __attribute__((ext_vector_type(4))) unsigned int v4u;
typedef __attribute__((ext_vector_type(8))) int          v8i;
typedef __attribute__((ext_vector_type(4))) int          v4i;

// Issue one 1-D TDM copy: copies tileElems 4-byte elements from the global address gptr to the LDS address lds_off.
// tensor_dim0 = nElemRemain so reads past the array tail return zero (tail tile).
__device__ __forceinline__ void tdm_load_1d(uint32_t lds_off, const void* gptr,
                                            uint32_t nElemRemain, uint32_t tileElems) {
    uint64_t ga = (uint64_t)(uintptr_t)gptr;

    // D# group 0 (128b): count=1 | lds_addr | global_addr[56:0] | type=2
    v4u g0;
    g0.x = 0x1u;
    g0.y = lds_off;
    g0.z = (uint32_t)ga;
    g0.w = ((uint32_t)(ga >> 32) & 0x1FFFFFFu) | (2u << 30);

    // D# group 1 (256b): wg_mask=0, data_size=2 (4B), no barrier/iterate/pad.
    v8i g1;
    g1[0] = (int)(2u << 16);                                     // data_size = 4 bytes
    g1[1] = (int)((nElemRemain & 0xFFFFu) << 16);                // tensor_dim0[15:0]
    g1[2] = (int)(((nElemRemain >> 16) & 0xFFFFu) | (1u << 16)); // td0[31:16] | tensor_dim1=1
    g1[3] = (int)(tileElems << 16);                              // tile_dim0
    g1[4] = 0;                                                   // tile_dim1=0, tile_dim2=0
    g1[5] = (int)nElemRemain;                                    // tensor_dim0_stride (unused, 1 row)
    g1[6] = 0;
    g1[7] = 0;

    v4i z4 = {0, 0, 0, 0};                    // groups 2/3 unused (<=2D tensor)
    v8i z8 = {0, 0, 0, 0, 0, 0, 0, 0};        // trailing descriptor payload unused
    __builtin_amdgcn_tensor_load_to_lds(g0, g1, z4, z4, z8, 0);
}

__global__ void __launch_bounds__(256)
zero_out_kernel(float* __restrict__ out, size_t n) {
    size_t i      = (size_t)blockIdx.x * blockDim.x + threadIdx.x;
    size_t stride = (size_t)gridDim.x * blockDim.x;
    size_t n4     = n >> 2;
    float4* o4 = reinterpret_cast<float4*>(out);
    const float4 z = make_float4(0.f, 0.f, 0.f, 0.f);
    for (size_t k = i; k < n4; k += stride) o4[k] = z;
    for (size_t k = (n4 << 2) + i; k < n; k += stride) out[k] = 0.f;
}

// grid.x = tiles over edges, grid.y = etype (0: buy user->item, 1: bought item->user)
__global__ void __launch_bounds__(256)
gnn_scatter_kernel(const float* __restrict__ h_user,
                   const float* __restrict__ h_item,
                   const int*   __restrict__ buy_src,
                   const int*   __restrict__ buy_dst,
                   const float* __restrict__ w_buy,
                   const int*   __restrict__ bought_src,
                   const int*   __restrict__ bought_dst,
                   const float* __restrict__ w_bought,
                   float* __restrict__ out_user,
                   float* __restrict__ out_item,
                   int nE_buy, int nE_bought) {
    __shared__ int   s_src[TILE];
    __shared__ int   s_dst[TILE];
    __shared__ float s_w[TILE];

    const int etype     = blockIdx.y;
    const int nE        = (etype == 0) ? nE_buy : nE_bought;
    const int tileStart = blockIdx.x * TILE;
    if (tileStart >= nE) return;

    const int*   srcArr = (etype == 0) ? buy_src    : bought_src;
    const int*   dstArr = (etype == 0) ? buy_dst    : bought_dst;
    const float* wArr   = (etype == 0) ? w_buy      : w_bought;
    const float* hsrc   = (etype == 0) ? h_user     : h_item;
    float*       outp   = (etype == 0) ? out_item   : out_user;

    const uint32_t remain = (uint32_t)(nE - tileStart);

    // Wave 0 stages the edge tile into LDS via the Tensor Data Mover.
    if (threadIdx.x < 32) {
        tdm_load_1d((uint32_t)(uintptr_t)(void*)s_src, srcArr + tileStart, remain, TILE);
        tdm_load_1d((uint32_t)(uintptr_t)(void*)s_dst, dstArr + tileStart, remain, TILE);
        tdm_load_1d((uint32_t)(uintptr_t)(void*)s_w,   wArr   + tileStart, remain, TILE);
        __builtin_amdgcn_s_wait_tensorcnt(0);
    }
    __syncthreads();

    // 16 lanes per edge; each lane owns one float4 (16 B) of the 64-float row.
    const int eg   = threadIdx.x >> 4;   // 0..15: edge slot within a pass
    const int lane = threadIdx.x & 15;   // 0..15: float4 index within row

#pragma unroll 4
    for (int p = 0; p < TILE / 16; ++p) {
        const int e = p * 16 + eg;            // edge index within tile
        if (tileStart + e < nE) {
            const int   s = s_src[e];
            const int   d = s_dst[e];
            const float w = s_w[e];
            const float4 v =
                reinterpret_cast<const float4*>(hsrc + (size_t)s * 64)[lane];
            float* o = outp + (size_t)d * 64 + lane * 4;
            atomicAdd(o + 0, v.x * w);
            atomicAdd(o + 1, v.y * w);
            atomicAdd(o + 2, v.z * w);
            atomicAdd(o + 3, v.w * w);
        }
    }
}

extern "C" void kernel_launch(void* const* d_in, const int* in_sizes, int n_in,
                              void* d_out, int out_size, void* d_ws, size_t ws_size,
                              hipStream_t stream) {
    const float* h_user     = (const float*)d_in[0];
    const float* h_item     = (const float*)d_in[1];
    const int*   buy_src    = (const int*)  d_in[2];
    const int*   buy_dst    = (const int*)  d_in[3];
    const float* w_buy      = (const float*)d_in[4];
    const int*   bought_src = (const int*)  d_in[5];
    const int*   bought_dst = (const int*)  d_in[6];
    const float* w_bought   = (const float*)d_in[7];

    const int D         = 64;
    const int n_user    = in_sizes[0] / D;
    const int nE_buy    = in_sizes[2];
    const int nE_bought = in_sizes[5];

    float* out_user = (float*)d_out;                       // [n_user, 64]
    float* out_item = out_user + (size_t)n_user * D;       // [n_item, 64]

    // 1) Zero the (poisoned) output.
    zero_out_kernel<<<2048, 256, 0, stream>>>((float*)d_out, (size_t)out_size);

    // 2) Fused dual-etype gather/scale/scatter-add.
    const int maxE  = (nE_buy > nE_bought) ? nE_buy : nE_bought;
    dim3 grid((maxE + TILE - 1) / TILE, 2, 1);
    gnn_scatter_kernel<<<grid, 256, 0, stream>>>(
        h_user, h_item,
        buy_src, buy_dst, w_buy,
        bought_src, bought_dst, w_bought,
        out_user, out_item,
        nE_buy, nE_bought);
}